// GCNEncoder_85538568667513
// MI455X (gfx1250) — compile-verified
//
#include <hip/hip_runtime.h>

// GCN encoder for MI455X (gfx1250).
// - layers 2/3 + node-sum collapse algebraically to one weighted row reduction
// - layer-1 GEMM (50000x128 @ 128x96) runs on V_WMMA_F32_16X16X4_F32
// - edge scatter-add uses hardware GLOBAL_ATOMIC_ADD_F32 via unsafeAtomicAdd
//   (executed at L2; agg/hpre [19MB each] are L2-resident on the 192MB L2)

typedef __attribute__((ext_vector_type(2))) float v2f;
typedef __attribute__((ext_vector_type(8))) float v8f;

#define FIN  128   // input features
#define FH   96    // hidden features
#define FOUT 64    // output features

// hardware fp32 atomic add (GLOBAL_ATOMIC_ADD_F32), not a CAS loop
__device__ __forceinline__ void atomAddF(float* p, float v) {
    unsafeAtomicAdd(p, v);
}

// ---------------- utility ----------------
__global__ void zero_kernel(float* __restrict__ p, int n) {
    int i = blockIdx.x * blockDim.x + threadIdx.x;
    if (i < n) p[i] = 0.0f;
}

// deg[dst[e]] += 1  (integer hardware atomics)
__global__ void deg_kernel(const int* __restrict__ dst, int E, unsigned* __restrict__ deg) {
    int e = blockIdx.x * blockDim.x + threadIdx.x;
    if (e < E) atomicAdd(&deg[dst[e]], 1u);
}

// in-place: buffer holds uint counts, becomes float dinv = (deg+1)^-0.5
__global__ void dinv_kernel(float* __restrict__ degbuf, int N) {
    int i = blockIdx.x * blockDim.x + threadIdx.x;
    if (i < N) {
        unsigned c = ((const unsigned*)degbuf)[i];
        degbuf[i] = rsqrtf((float)c + 1.0f);   // self-loop => deg >= 1
    }
}

// s[src[e]] += dinv[dst[e]]
__global__ void s_kernel(const int* __restrict__ src, const int* __restrict__ dst,
                         int E, const float* __restrict__ dinv, float* __restrict__ s) {
    int e = blockIdx.x * blockDim.x + threadIdx.x;
    if (e < E) atomAddF(&s[src[e]], dinv[dst[e]]);
}

// ---------------- layer-1 GEMM: hpre = X(50000x128) @ W1(128x96) ----------------
// One block = 16 output rows; 6 waves, wave w owns N-tile [16w, 16w+16).
// W1 (48KB) and the 16x128 A tile (8KB) are staged in LDS.
__global__ __launch_bounds__(192) void gemm_wmma_kernel(
        const float* __restrict__ X, const float* __restrict__ W1,
        float* __restrict__ H, int Nnodes) {
    __shared__ float ldsW[FIN * FH];   // 48 KB
    __shared__ float ldsA[16 * FIN];   //  8 KB

    const int tid = threadIdx.x;
    const int m0  = blockIdx.x * 16;

    for (int i = tid; i < FIN * FH; i += 192) ldsW[i] = W1[i];
    for (int i = tid; i < 16 * FIN; i += 192) {
        int r = i >> 7;            // i / 128
        int c = i & 127;           // i % 128
        int row = m0 + r; if (row >= Nnodes) row = Nnodes - 1;
        ldsA[i] = X[(size_t)row * FIN + c];
    }
    __syncthreads();

    const int wave = tid >> 5;     // 0..5
    const int lane = tid & 31;
    const int half = lane >> 4;    // 0: K low pair / M 0..7, 1: K high pair / M 8..15
    const int l    = lane & 15;
    const int n0   = wave * 16;

    v8f acc = {};
    for (int k = 0; k < FIN; k += 4) {
        const int kk = k + 2 * half;
        v2f a, b;
        // A 16x4 f32 layout: lanes 0-15 hold K=0,1 ; lanes 16-31 hold K=2,3 (M = l)
        a.x = ldsA[l * FIN + kk];
        a.y = ldsA[l * FIN + kk + 1];
        // B 4x16 f32 layout (mirrored): lanes 0-15 rows K=0,1 ; lanes 16-31 rows K=2,3 (N = l)
        b.x = ldsW[kk * FH + n0 + l];
        b.y = ldsW[(kk + 1) * FH + n0 + l];
        acc = __builtin_amdgcn_wmma_f32_16x16x4_f32(
                /*neg_a=*/false, a, /*neg_b=*/false, b,
                /*c_mod=*/(short)0, acc, /*reuse_a=*/false, /*reuse_b=*/false);
    }

    // C/D f32 16x16: VGPR r, lanes 0-15 -> M=r, lanes 16-31 -> M=8+r, N=l
#pragma unroll
    for (int r = 0; r < 8; ++r) {
        int row = m0 + r + 8 * half;
        if (row < Nnodes) H[(size_t)row * FH + n0 + l] = acc[r];
    }
}

// ---------------- edge aggregation: agg[dst] += norm * hpre[src] ----------------
// one wave per edge; each lane handles 3 of the 96 columns (coalesced row access)
__global__ void aggregate_kernel(const int* __restrict__ src, const int* __restrict__ dst,
                                 int E, const float* __restrict__ dinv,
                                 const float* __restrict__ H, float* __restrict__ agg) {
    int gid  = blockIdx.x * blockDim.x + threadIdx.x;
    int e    = gid >> 5;
    int lane = gid & 31;
    if (e >= E) return;
    int si = src[e], di = dst[e];
    float norm = dinv[si] * dinv[di];
    const float* hrow = H   + (size_t)si * FH;
    float*       arow = agg + (size_t)di * FH;
    atomAddF(&arow[lane],      hrow[lane]      * norm);
    atomAddF(&arow[lane + 32], hrow[lane + 32] * norm);
    atomAddF(&arow[lane + 64], hrow[lane + 64] * norm);
}

// ---------------- fused self-loop + bias + ReLU (in place into agg -> h) --------
__global__ void relu_kernel(float* __restrict__ agg, const float* __restrict__ hpre,
                            const float* __restrict__ dinv, const float* __restrict__ b1,
                            int total) {
    int idx = blockIdx.x * blockDim.x + threadIdx.x;
    if (idx >= total) return;
    int i = idx / FH;
    int c = idx - i * FH;
    float d = dinv[i];
    float v = agg[idx] + d * d * hpre[idx] + b1[c];
    agg[idx] = fmaxf(v, 0.0f);
}

// ---------------- weighted reduction: g[c] = sum_i w_i * h[i,c] -----------------
// w_i = dinv_i * (s_i + dinv_i);  96 threads per block, 128 nodes per block
__global__ __launch_bounds__(96) void reduce_kernel(
        const float* __restrict__ h, const float* __restrict__ dinv,
        const float* __restrict__ s, int N, float* __restrict__ g) {
    const int c  = threadIdx.x;                 // 0..95
    const int i0 = blockIdx.x * 128;
    int i1 = i0 + 128; if (i1 > N) i1 = N;
    float acc = 0.0f;
    for (int i = i0; i < i1; ++i) {
        float d = dinv[i];
        float w = d * (s[i] + d);
        acc += w * h[(size_t)i * FH + c];
    }
    atomAddF(&g[c], acc);
}

// ---------------- head: out = [g@Wmu + N*bmu, g@Wlv + N*blv] --------------------
__global__ void head_kernel(const float* __restrict__ g,
                            const float* __restrict__ Wmu, const float* __restrict__ bmu,
                            const float* __restrict__ Wlv, const float* __restrict__ blv,
                            float Nf, float* __restrict__ out) {
    int c = threadIdx.x;                        // 0..63
    float mu = 0.0f, lv = 0.0f;
    for (int k = 0; k < FH; ++k) {
        float gv = g[k];
        mu += gv * Wmu[k * FOUT + c];
        lv += gv * Wlv[k * FOUT + c];
    }
    out[c]        = mu + Nf * bmu[c];
    out[FOUT + c] = lv + Nf * blv[c];
}

// ---------------- launch ----------------
extern "C" void kernel_launch(void* const* d_in, const int* in_sizes, int n_in,
                              void* d_out, int out_size, void* d_ws, size_t ws_size,
                              hipStream_t stream) {
    const float* x   = (const float*)d_in[0];
    const int*   ei  = (const int*)  d_in[1];
    const float* W1  = (const float*)d_in[2];
    const float* b1  = (const float*)d_in[3];
    const float* Wmu = (const float*)d_in[4];
    const float* bmu = (const float*)d_in[5];
    const float* Wlv = (const float*)d_in[6];
    const float* blv = (const float*)d_in[7];

    const int N = in_sizes[0] / FIN;   // 50000 nodes
    const int E = in_sizes[1] / 2;     // 800000 edges
    const int* src = ei;
    const int* dst = ei + E;

    // workspace layout (floats): deg/dinv[N] | s[N] | g[96 (+pad 32)] | hpre[N*96] | agg[N*96]
    float* ws   = (float*)d_ws;
    float* deg  = ws;                       // uint during counting, then dinv
    float* svec = ws + N;
    float* g    = ws + 2 * (size_t)N;
    float* hpre = ws + 2 * (size_t)N + 128;
    float* agg  = hpre + (size_t)N * FH;
    // total ws use: (2N + 128 + 2*N*96)*4 ~= 38.8 MB

    const int hz = N * FH;                  // 4.8M elements
    zero_kernel<<<(2 * N + 128 + 255) / 256, 256, 0, stream>>>(ws, 2 * N + 128);
    zero_kernel<<<(hz + 255) / 256, 256, 0, stream>>>(agg, hz);

    deg_kernel<<<(E + 255) / 256, 256, 0, stream>>>(dst, E, (unsigned*)deg);
    dinv_kernel<<<(N + 255) / 256, 256, 0, stream>>>(deg, N);
    s_kernel<<<(E + 255) / 256, 256, 0, stream>>>(src, dst, E, deg, svec);

    gemm_wmma_kernel<<<(N + 15) / 16, 192, 0, stream>>>(x, W1, hpre, N);

    aggregate_kernel<<<(E * 32 + 255) / 256, 256, 0, stream>>>(src, dst, E, deg, hpre, agg);
    relu_kernel<<<(hz + 255) / 256, 256, 0, stream>>>(agg, hpre, deg, b1, hz);
    reduce_kernel<<<(N + 127) / 128, 96, 0, stream>>>(agg, deg, svec, N, g);
    head_kernel<<<1, 64, 0, stream>>>(g, Wmu, bmu, Wlv, blv, (float)N, (float*)d_out);
}